// SRUCell_42915313221800
// MI455X (gfx1250) — compile-verified
//
#include <hip/hip_runtime.h>
#include <math.h>

typedef float v2f __attribute__((ext_vector_type(2)));
typedef float v8f __attribute__((ext_vector_type(8)));
typedef int   v4i_ __attribute__((ext_vector_type(4)));

#define N_IN   1024
#define N_OUT  1024
#define LSEQ   2048
#define BATCH  16
#define NPRIME 2048                 // 2*N_OUT compacted (x_tilde, forget) columns
#define MROWS  (LSEQ * BATCH)       // 32768 GEMM rows

#define BM 128
#define BN 256
#define BK 32
#define LDA 36     // 32 + 4 pad: {36*l mod 64} distinct & ==0 mod 4 for l=0..15
#define LDBP 544   // 256 cols * 2 (K-pair) + 32 pad: 544 mod 64 == 32 -> halves hit disjoint banks
#define BROWS 16   // B LDS rows per slab: 8 K-quads * 2 lane-halves

// ---- CDNA5 async global->LDS path (ASYNCcnt), guarded so the file always compiles
#if defined(__has_builtin)
#if __has_builtin(__builtin_amdgcn_global_load_async_to_lds_b128) && \
    __has_builtin(__builtin_amdgcn_s_wait_asynccnt)
#define USE_ASYNC_LDS 1
#endif
#endif
#ifndef USE_ASYNC_LDS
#define USE_ASYNC_LDS 0
#endif

__device__ __forceinline__ void cp16_to_lds(const float* __restrict__ g, float* l) {
#if USE_ASYNC_LDS
    // builtin expects plain (generic, non-const) v4i* pointers per hipcc diagnostic
    __builtin_amdgcn_global_load_async_to_lds_b128(
        (v4i_*)(void*)const_cast<float*>(g),
        (v4i_*)(void*)l,
        /*offset=*/0, /*cpol=*/0);
#else
    *(float4*)l = *(const float4*)g;
#endif
}

__device__ __forceinline__ void async_wait_all() {
#if USE_ASYNC_LDS
    __builtin_amdgcn_s_wait_asynccnt(0);
#endif
}

// ---------------- kernel 1: repack weight (N_IN x 3*N_OUT) -> pair-packed Wp
// Layout: Wp[((q*2+hl)*NPRIME + n)*2 + pos] = W[(4q+hl+2*pos)][3*(n>>1)+(n&1)]
// so a lane's B fragment {B[kk+hl][n], B[kk+hl+2][n]} is one contiguous float2.
__global__ __launch_bounds__(256) void sru_repack(const float* __restrict__ W,
                                                  float* __restrict__ Wp) {
    int o   = blockIdx.x * 256 + threadIdx.x;     // 0 .. N_IN*NPRIME-1
    int pos = o & 1;
    int n   = (o >> 1) & (NPRIME - 1);
    int R   = o >> 12;                            // q*2 + hl
    int q   = R >> 1, hl = R & 1;
    int k   = 4 * q + hl + 2 * pos;
    Wp[o] = W[(size_t)k * (3 * N_OUT) + 3 * (n >> 1) + (n & 1)];
}

// ---------------- kernel 2: fp32 WMMA GEMM, async double-buffered LDS
// X: (MROWS, N_IN) row-major; Wp: pair-packed (see above); U: (MROWS, NPRIME)
__global__ __launch_bounds__(256) void sru_gemm(const float* __restrict__ X,
                                                const float* __restrict__ Wp,
                                                const float* __restrict__ bias,
                                                float* __restrict__ U) {
    __shared__ float As[2][BM * LDA];
    __shared__ float Bs[2][BROWS * LDBP];

    const int tid  = threadIdx.x;
    const int wave = tid >> 5;
    const int lane = tid & 31;
    const int hl   = lane >> 4;      // wave32 lane half
    const int l16  = lane & 15;

    const int m0 = blockIdx.y * BM;
    const int n0 = blockIdx.x * BN;
    const int wRow = (wave & 1) * 64;   // 2 waves in M
    const int wCol = (wave >> 1) * 64;  // 4 waves in N

    v8f acc[4][4];
#pragma unroll
    for (int mt = 0; mt < 4; ++mt)
#pragma unroll
        for (int nt = 0; nt < 4; ++nt)
            acc[mt][nt] = (v8f){0.f, 0.f, 0.f, 0.f, 0.f, 0.f, 0.f, 0.f};

    // issue one slab's tile loads (A: 1024 16B chunks, B: 1024 16B chunks; 4+4 per thread)
    auto loadSlab = [&](int k0, int buf) {
#pragma unroll
        for (int i = 0; i < 4; ++i) {
            int c = tid + i * 256;
            int row = c >> 3, kq = (c & 7) << 2;
            cp16_to_lds(&X[(size_t)(m0 + row) * N_IN + k0 + kq],
                        &As[buf][row * LDA + kq]);
        }
#pragma unroll
        for (int i = 0; i < 4; ++i) {
            int c = tid + i * 256;
            int r = c >> 7, j = c & 127;              // r: 0..7, j: 0..127 (16B chunks)
            // global pair-packed row R = k0/2 + r, row length NPRIME*2 floats
            cp16_to_lds(&Wp[(size_t)((k0 >> 1) + r) * (NPRIME * 2) + n0 * 2 + j * 4],
                        &Bs[buf][r * LDBP + j * 4]);
        }
    };

    loadSlab(0, 0);

    for (int k0 = 0; k0 < N_IN; k0 += BK) {
        const int buf = (k0 >> 5) & 1;
        async_wait_all();            // this wave's async copies into `buf` landed
        __syncthreads();             // everyone's copies landed
        if (k0 + BK < N_IN) loadSlab(k0 + BK, buf ^ 1);

#pragma unroll
        for (int kk = 0; kk < BK; kk += 4) {
            const int qq = kk >> 2;
            v2f a[4], b[4];
#pragma unroll
            for (int mt = 0; mt < 4; ++mt)
                a[mt] = *(const v2f*)&As[buf][(wRow + mt * 16 + l16) * LDA + kk + hl * 2];
#pragma unroll
            for (int nt = 0; nt < 4; ++nt)
                b[nt] = *(const v2f*)&Bs[buf][(qq * 2 + hl) * LDBP + (wCol + nt * 16 + l16) * 2];
#pragma unroll
            for (int mt = 0; mt < 4; ++mt)
#pragma unroll
                for (int nt = 0; nt < 4; ++nt)
                    acc[mt][nt] = __builtin_amdgcn_wmma_f32_16x16x4_f32(
                        /*neg_a=*/false, a[mt],
                        /*neg_b=*/false, b[nt],
                        /*c_mod=*/(short)0, acc[mt][nt],
                        /*reuse_a=*/false, /*reuse_b=*/false);
        }
        __syncthreads();             // done reading `buf` before it's refilled
    }

    // ---- epilogue: C/D layout VGPR r -> rows r (lanes0-15) / r+8 (lanes16-31)
    // even col = x_tilde (raw); odd col = forget gate: sigmoid(u + bias[d])
#pragma unroll
    for (int mt = 0; mt < 4; ++mt) {
#pragma unroll
        for (int nt = 0; nt < 4; ++nt) {
#pragma unroll
            for (int r = 0; r < 8; ++r) {
                int row = m0 + wRow + mt * 16 + r + hl * 8;
                int col = n0 + wCol + nt * 16 + l16;
                float v = acc[mt][nt][r];
                if (col & 1) {
                    v += bias[col >> 1];
                    v = 1.0f / (1.0f + __expf(-v));
                }
                U[(size_t)row * NPRIME + col] = v;
            }
        }
    }
}

// ---------------- kernel 3: sequential SRU recurrence + tanh
// c_t = (c_{t-1} - x~_t) * f_t + x~_t ; h_t = tanh(c_t)
__global__ __launch_bounds__(256) void sru_scan(const float* __restrict__ U,
                                                const float* __restrict__ c0,
                                                float* __restrict__ H,
                                                float* __restrict__ Cf) {
    int id = blockIdx.x * 256 + threadIdx.x;   // 0 .. BATCH*N_OUT-1
    int b  = id >> 10;                         // / N_OUT
    int d  = id & (N_OUT - 1);

    float c = c0[b * N_OUT + d];
    const float2* __restrict__ Up = (const float2*)U;   // (x_tilde, f) pairs

#pragma unroll 4
    for (int t = 0; t < LSEQ; ++t) {
        size_t rc = (size_t)(t * BATCH + b) * N_OUT + d;
        float2 p = Up[rc];
        c = (c - p.x) * p.y + p.x;
        H[rc] = tanhf(c);
    }
    Cf[b * N_OUT + d] = c;
}

extern "C" void kernel_launch(void* const* d_in, const int* in_sizes, int n_in,
                              void* d_out, int out_size, void* d_ws, size_t ws_size,
                              hipStream_t stream) {
    const float* x    = (const float*)d_in[0];   // (L, B, N_IN)
    const float* c0   = (const float*)d_in[1];   // (B, N_OUT)
    const float* W    = (const float*)d_in[2];   // (N_IN, 3*N_OUT)
    const float* bias = (const float*)d_in[3];   // (2*N_OUT,), first N_OUT used

    float* Wp = (float*)d_ws;                                                  // 8 MB
    float* U  = (float*)((char*)d_ws + (size_t)N_IN * NPRIME * sizeof(float)); // 256 MB

    float* H  = (float*)d_out;                         // (L, B, N_OUT)
    float* Cf = H + (size_t)LSEQ * BATCH * N_OUT;      // (B, N_OUT)

    sru_repack<<<(N_IN * NPRIME) / 256, 256, 0, stream>>>(W, Wp);

    dim3 grid(NPRIME / BN, MROWS / BM);                // (8, 256)
    sru_gemm<<<grid, 256, 0, stream>>>(x, Wp, bias, U);

    sru_scan<<<(BATCH * N_OUT) / 256, 256, 0, stream>>>(U, c0, H, Cf);
}